// SinkhornAlgorithm_46763603918987
// MI455X (gfx1250) — compile-verified
//
#include <hip/hip_runtime.h>
#include <hip/hip_bf16.h>
#include <cstdint>
#include <cstddef>
#include <cmath>

// ---------------------------------------------------------------------------
// Sinkhorn (B=16, N=M=2048, fp32) for MI455X / gfx1250.
//
// u_new[b,i] = eps*log(p+1e-8) - eps*ln( sum_j exp((v[b,j]-C[b,i,j])/eps) )
// v_new[b,j] = eps*log(q+1e-8) - eps*ln( sum_i exp((u[b,i]-C[b,i,j])/eps) )
// pi = exp((-C + u_i + v_j)/eps)
//
// Memory-bound: 200 streaming passes over 256 MiB of C (~52 GB -> ~2.2 ms at
// 23.3 TB/s; L2 at 192 MB keeps most of C resident). Row-pass reductions are
// done on the matrix unit with V_WMMA_F32_16X16X4_F32 (B = ones), exp on the
// transcendental pipe (v_exp_f32), so VALU stays mostly free for addressing.
// ---------------------------------------------------------------------------

typedef float v2f __attribute__((ext_vector_type(2)));
typedef float v8f __attribute__((ext_vector_type(8)));

#define EPS_SINK   0.1f
#define LOG2E      1.44269504088896340736f
#define LN2        0.69314718055994530942f
#define EXP_SCALE  (LOG2E / EPS_SINK)          // exp(x/eps) == exp2(x*EXP_SCALE)
#define ITERS      100
#define RSPLIT     8                           // row split for the column pass

static __device__ __forceinline__ float fexp2(float x) {
#if defined(__HIP_DEVICE_COMPILE__) && __has_builtin(__builtin_amdgcn_exp2f)
    return __builtin_amdgcn_exp2f(x);          // v_exp_f32
#else
    return exp2f(x);
#endif
}
static __device__ __forceinline__ float flog2(float x) {
#if defined(__HIP_DEVICE_COMPILE__) && __has_builtin(__builtin_amdgcn_logf)
    return __builtin_amdgcn_logf(x);           // v_log_f32
#else
    return log2f(x);
#endif
}

// ------------------------- init: elp/elq/v0 --------------------------------
__global__ void sink_init(const float* __restrict__ p, const float* __restrict__ q,
                          float* __restrict__ elp, float* __restrict__ elq,
                          float* __restrict__ v, int np, int nq) {
    int t = blockIdx.x * blockDim.x + threadIdx.x;
    if (t < np) elp[t] = EPS_SINK * (LN2 * flog2(p[t] + 1e-8f));
    if (t < nq) {
        elq[t] = EPS_SINK * (LN2 * flog2(q[t] + 1e-8f));
        v[t]   = 0.0f;
    }
}

// ------------------------- row pass (u update) -----------------------------
// Block = 128 threads (4 wave32), grid = (N/16, B). Each block owns 16 rows
// of one batch and sweeps all M columns in 16x128 LDS tiles.
// WMMA A layout (16x4 f32): lane L holds row (L&15), K-pair 2*(L>>4).
#define TW         128
#define LDS_STRIDE (TW + 4)   // 132: rows map to distinct bank groups (4*r)

__global__ void __launch_bounds__(128)
sink_row_pass(const float* __restrict__ C, const float* __restrict__ v,
              const float* __restrict__ elp, float* __restrict__ u,
              int N, int M) {
    __shared__ float tile[16 * LDS_STRIDE];
    __shared__ float vt[TW];
    __shared__ float wsum[4][16];

    const int b    = blockIdx.y;
    const int row0 = blockIdx.x * 16;
    const int t    = threadIdx.x;      // 0..127
    const int lane = t & 31;
    const int wave = t >> 5;           // 0..3

    const float* Cb = C + ((size_t)b * N + row0) * (size_t)M;
    const float* vb = v + (size_t)b * M;

    // A-fragment mapping for this lane
    const int arow = lane & 15;
    const int koff = (lane >> 4) * 2;

    // cooperative tile-load mapping: 8 threads per row, 16 floats each
    const int loadRow = t >> 3;
    const int loadCol = (t & 7) * 16;

    v8f acc = {};                       // 16 row-sums, accumulated by WMMA
    const v2f bones = {1.0f, 1.0f};     // B = ones (4x16) -> row reduction

    for (int tb = 0; tb < M; tb += TW) {
        // ---- stage 16x128 tile of C into LDS (b128 in, b128 out) ----
        const float* src = Cb + (size_t)loadRow * M + tb + loadCol;
        float* dst = &tile[loadRow * LDS_STRIDE + loadCol];
#pragma unroll
        for (int k = 0; k < 4; ++k) {
            float4 val = *(const float4*)(src + k * 4);
            *(float4*)(dst + k * 4) = val;
        }
        vt[t & (TW - 1)] = vb[tb + (t & (TW - 1))];
        if (tb + TW < M)                 // prefetch next tile while we compute
            __builtin_prefetch(src + TW, 0, 0);   // global_prefetch_b8
        __syncthreads();

        // ---- each wave reduces its 32 columns: 8 chunks of K=4 ----
#pragma unroll
        for (int c = 0; c < 8; ++c) {
            const int jloc = wave * 32 + c * 4 + koff;
            v2f cv = *(const v2f*)(&tile[arow * LDS_STRIDE + jloc]); // ds_load_b64
            v2f a;
            a.x = fexp2((vt[jloc]     - cv.x) * EXP_SCALE);
            a.y = fexp2((vt[jloc + 1] - cv.y) * EXP_SCALE);
            // D = A(16x4) * ones(4x16) + C : every column of D = row sums
            acc = __builtin_amdgcn_wmma_f32_16x16x4_f32(
                false, a, false, bones, (short)0, acc, false, false);
        }
        __syncthreads();
    }

    // acc: vgpr m / lanes 0-15 -> row m ; lanes 16-31 -> row m+8 (all n equal)
    if (lane == 0) {
#pragma unroll
        for (int m = 0; m < 8; ++m) wsum[wave][m] = acc[m];
    } else if (lane == 16) {
#pragma unroll
        for (int m = 0; m < 8; ++m) wsum[wave][8 + m] = acc[m];
    }
    __syncthreads();

    if (t < 16) {
        float s = wsum[0][t] + wsum[1][t] + wsum[2][t] + wsum[3][t];
        const size_t gi = (size_t)b * N + row0 + t;
        u[gi] = elp[gi] - EPS_SINK * (LN2 * flog2(s));
    }
}

// ------------------------- column pass (v update) --------------------------
// lane = column (coalesced b32 stream over rows); rows split RSPLIT ways for
// occupancy; partial sums land in workspace, combined by sink_finish_v.
__global__ void __launch_bounds__(256)
sink_col_pass(const float* __restrict__ C, const float* __restrict__ u,
              float* __restrict__ partial, int N, int M, int B, int rowsPer) {
    const int b = blockIdx.z;
    const int r = blockIdx.y;
    const int j = blockIdx.x * blockDim.x + threadIdx.x;
    const int i0 = r * rowsPer;

    const float* ub = u + (size_t)b * N + i0;
    const float* cp = C + ((size_t)b * N + i0) * (size_t)M + j;

    float acc = 0.0f;
#pragma unroll 4
    for (int i = 0; i < rowsPer; ++i) {
        float cu = ub[i];                       // uniform -> scalar load
        acc += fexp2((cu - cp[(size_t)i * M]) * EXP_SCALE);
    }
    partial[((size_t)r * B + b) * M + j] = acc;
}

__global__ void __launch_bounds__(256)
sink_finish_v(const float* __restrict__ partial, const float* __restrict__ elq,
              float* __restrict__ v, int M, int B) {
    const int b = blockIdx.y;
    const int j = blockIdx.x * blockDim.x + threadIdx.x;
    float s = 0.0f;
#pragma unroll
    for (int r = 0; r < RSPLIT; ++r)
        s += partial[((size_t)r * B + b) * M + j];
    const size_t gi = (size_t)b * M + j;
    v[gi] = elq[gi] - EPS_SINK * (LN2 * flog2(s));
}

// ------------------------- final transport plan ----------------------------
__global__ void __launch_bounds__(256)
sink_pi(const float* __restrict__ C, const float* __restrict__ u,
        const float* __restrict__ v, float* __restrict__ out, int N, int M) {
    const int b = blockIdx.z;
    const int i = blockIdx.y;
    const int j = (blockIdx.x * blockDim.x + threadIdx.x) * 4;

    const size_t base = ((size_t)b * N + i) * (size_t)M + j;
    const float ui = u[(size_t)b * N + i];
    float4 c4 = *(const float4*)(C + base);
    float4 v4 = *(const float4*)(v + (size_t)b * M + j);
    float4 o;
    o.x = fexp2((ui + v4.x - c4.x) * EXP_SCALE);
    o.y = fexp2((ui + v4.y - c4.y) * EXP_SCALE);
    o.z = fexp2((ui + v4.z - c4.z) * EXP_SCALE);
    o.w = fexp2((ui + v4.w - c4.w) * EXP_SCALE);
    *(float4*)(out + base) = o;
}

// ---------------------------------------------------------------------------
extern "C" void kernel_launch(void* const* d_in, const int* in_sizes, int n_in,
                              void* d_out, int out_size, void* d_ws, size_t ws_size,
                              hipStream_t stream) {
    const float* p = (const float*)d_in[0];
    const float* q = (const float*)d_in[1];
    const float* C = (const float*)d_in[2];
    float* out = (float*)d_out;

    const int s0 = in_sizes[0];          // B*N
    const int s1 = in_sizes[1];          // B*M
    const long long s2 = in_sizes[2];    // B*N*M
    const int M = (int)(s2 / s0);
    const int N = (int)(s2 / s1);
    const int B = s0 / N;

    // Workspace layout (floats): u[s0] | v[s1] | elp[s0] | elq[s1] | partial[RSPLIT*s1]
    float* u       = (float*)d_ws;
    float* v       = u + s0;
    float* elp     = v + s1;
    float* elq     = elp + s0;
    float* partial = elq + s1;
    // requires ws_size >= (2*s0 + 2*s1 + RSPLIT*s1) * 4  (~1.5 MB here)

    const int mx = (s0 > s1 ? s0 : s1);
    sink_init<<<(mx + 255) / 256, 256, 0, stream>>>(p, q, elp, elq, v, s0, s1);

    const int rowsPer = N / RSPLIT;
    for (int it = 0; it < ITERS; ++it) {
        sink_row_pass<<<dim3(N / 16, B), 128, 0, stream>>>(C, v, elp, u, N, M);
        sink_col_pass<<<dim3(M / 256, RSPLIT, B), 256, 0, stream>>>(C, u, partial, N, M, B, rowsPer);
        sink_finish_v<<<dim3(M / 256, B), 256, 0, stream>>>(partial, elq, v, M, B);
    }
    sink_pi<<<dim3(M / 1024, N, B), 256, 0, stream>>>(C, u, v, out, N, M);
}